// BraileInferenceImpl_13683765805358
// MI455X (gfx1250) — compile-verified
//
#include <hip/hip_runtime.h>
#include <hip/hip_bf16.h>
#include <stdint.h>

// ---------------------------------------------------------------------------
// CDNA5 async global->LDS copy (ASYNCcnt path). Builtin if present, else ISA
// inline asm (mnemonic per CDNA5 ISA opcode 98 / CDNA5 ISA).
// Builtin signature (from hipcc diagnostic): first param is int4* in AS1
// (printed as "__device__"), so pass v4i-typed AS-qualified pointers.
// ---------------------------------------------------------------------------
#if defined(__has_builtin)
#  if __has_builtin(__builtin_amdgcn_global_load_async_to_lds_b128)
#    define HAVE_ASYNC_BUILTIN 1
#  endif
#  if __has_builtin(__builtin_amdgcn_s_wait_asynccnt)
#    define HAVE_WAIT_BUILTIN 1
#  endif
#endif

typedef int v4i __attribute__((vector_size(4 * sizeof(int))));
typedef __attribute__((address_space(1))) v4i* v4i_g;   // global (AS1)
typedef __attribute__((address_space(3))) v4i* v4i_l;   // LDS (AS3)

__device__ __forceinline__ void async_copy_b128(const float* gp, float* lp) {
#if defined(HAVE_ASYNC_BUILTIN)
  __builtin_amdgcn_global_load_async_to_lds_b128(
      (v4i_g)(uintptr_t)gp,
      (v4i_l)(unsigned)(uintptr_t)lp,
      0, 0);
#else
  unsigned lds32 = (unsigned)(uintptr_t)lp;            // low 32 bits = LDS offset
  unsigned long long ga = (unsigned long long)(uintptr_t)gp;
  asm volatile("global_load_async_to_lds_b128 %0, %1, off"
               :: "v"(lds32), "v"(ga) : "memory");
#endif
}

__device__ __forceinline__ void wait_async0() {
#if defined(HAVE_WAIT_BUILTIN)
  __builtin_amdgcn_s_wait_asynccnt(0);
#else
  asm volatile("s_wait_asynccnt 0" ::: "memory");
#endif
}

#define NEG_INF (-__builtin_inff())
#define MAX_DET 1000
#define NMS_THRESH 0.02f
#define CLS_THRESH 0.3f

// ---------------------------------------------------------------------------
// Pass 1: per-anchor class max/argmax (via async-staged LDS tile), sigmoid,
// threshold mask, box decode, area. 128 threads = 4 waves; each wave stages
// its 32 anchors' 64 logits (8 KB) into LDS with padded rows (stride 68 words
// -> rows stay 16B-aligned, removes the bank-conflict of stride 64).
// ---------------------------------------------------------------------------
__global__ __launch_bounds__(128) void decode_score_kernel(
    const float* __restrict__ loc, const float* __restrict__ cls,
    const float* __restrict__ anc, float* __restrict__ boxes,
    float* __restrict__ areas, float* __restrict__ scores,
    int* __restrict__ labels, float* __restrict__ masked, int A) {
  __shared__ float stage[4 * 32 * 68];   // 34,816 B

  const int tid = threadIdx.x;
  const int wv = tid >> 5;               // wave32
  const int ln = tid & 31;
  const int a0 = blockIdx.x * 128 + wv * 32;   // first anchor of this wave

  float* sbase = &stage[wv * (32 * 68)];
  const float* gbase = cls + (size_t)a0 * 64;

  // 16 async B128 copies per lane: wave moves 32 anchors x 64 f32 = 8 KB.
#pragma unroll
  for (int k = 0; k < 16; ++k) {
    int c = (k << 5) + ln;               // 16B chunk id within wave tile, 0..511
    int a = c >> 4;                      // anchor-in-wave 0..31
    int j = (c & 15) << 2;               // word offset within row
    const float* gp = ((a0 + a) < A) ? (gbase + ((size_t)c << 2)) : cls; // clamp OOB
    float* lp = sbase + a * 68 + j;      // padded row
    async_copy_b128(gp, lp);
  }
  wait_async0();
  __syncthreads();                       // ordering for LDS reads

  // Lane ln reduces its own anchor's 64 logits. First-max (lowest index) tie
  // semantics to match jnp.argmax.
  float m = NEG_INF;
  int mi = 0;
  const float* row = sbase + ln * 68;
#pragma unroll
  for (int j = 0; j < 64; j += 4) {
    float4 v = *reinterpret_cast<const float4*>(row + j);
    if (v.x > m) { m = v.x; mi = j + 0; }
    if (v.y > m) { m = v.y; mi = j + 1; }
    if (v.z > m) { m = v.z; mi = j + 2; }
    if (v.w > m) { m = v.w; mi = j + 3; }
  }

  const int a = a0 + ln;
  if (a < A) {
    float4 l  = *reinterpret_cast<const float4*>(loc + (size_t)a * 4);
    float4 an = *reinterpret_cast<const float4*>(anc + (size_t)a * 4);
    float cx = l.x * an.z + an.x;
    float cy = l.y * an.w + an.y;
    float w  = expf(l.z) * an.z;
    float h  = expf(l.w) * an.w;
    float x1 = cx - 0.5f * w, y1 = cy - 0.5f * h;
    float x2 = cx + 0.5f * w, y2 = cy + 0.5f * h;
    *reinterpret_cast<float4*>(boxes + (size_t)a * 4) = make_float4(x1, y1, x2, y2);
    areas[a] = (x2 - x1) * (y2 - y1);
    float s = 1.0f / (1.0f + expf(-m));
    scores[a] = s;
    labels[a] = mi;
    masked[a] = (s > CLS_THRESH) ? s : NEG_INF;
  }
}

// ---------------------------------------------------------------------------
// Pass 2: greedy NMS, 1000 dependent iterations. One persistent 1024-thread
// workgroup; hot data (scores 1.2MB, boxes 4.9MB) is L2-resident. Masked
// scores accessed volatile (bypass WGP-local caching across waves).
// ---------------------------------------------------------------------------
__global__ __launch_bounds__(1024) void nms_kernel(
    const float* __restrict__ boxes, const float* __restrict__ areas,
    const float* __restrict__ scores, const int* __restrict__ labels,
    float* masked_raw, float* __restrict__ out, int A) {
  __shared__ float sv[1024];
  __shared__ int   si[1024];
  __shared__ float cb[5];      // chosen x1,y1,x2,y2,area
  __shared__ int   cok;        // chosen valid flag

  volatile float* ms = masked_raw;
  const int tid = threadIdx.x;

  for (int it = 0; it < MAX_DET; ++it) {
    // ---- global argmax over surviving masked scores ----
    float m = NEG_INF;
    int mi = 0x7fffffff;
    for (int j = tid; j < A; j += 1024) {
      __builtin_prefetch(masked_raw + j + 4096, 0, 0);  // global_prefetch_b8
      float v = ms[j];
      if (v > m) { m = v; mi = j; }    // strictly-greater: keeps lowest index
    }
    sv[tid] = m; si[tid] = mi;
    __syncthreads();
    for (int s = 512; s > 0; s >>= 1) {
      if (tid < s) {
        float vo = sv[tid + s]; int io = si[tid + s];
        if (vo > sv[tid] || (vo == sv[tid] && io < si[tid])) {
          sv[tid] = vo; si[tid] = io;
        }
      }
      __syncthreads();
    }

    if (tid == 0) {
      int j = si[0];
      float val = sv[0];
      int ok = (val > NEG_INF) ? 1 : 0;
      cok = ok;
      if (ok) {
        float4 b = *reinterpret_cast<const float4*>(boxes + (size_t)j * 4);
        cb[0] = b.x; cb[1] = b.y; cb[2] = b.z; cb[3] = b.w; cb[4] = areas[j];
        out[it * 4 + 0] = b.x; out[it * 4 + 1] = b.y;
        out[it * 4 + 2] = b.z; out[it * 4 + 3] = b.w;
        out[4 * MAX_DET + it] = (float)labels[j];
        out[5 * MAX_DET + it] = scores[j];
        out[6 * MAX_DET + it] = 1.0f;
        ms[j] = NEG_INF;               // (arange == j) suppression
      } else {
        out[it * 4 + 0] = 0.0f; out[it * 4 + 1] = 0.0f;
        out[it * 4 + 2] = 0.0f; out[it * 4 + 3] = 0.0f;
        out[4 * MAX_DET + it] = -1.0f;
        out[5 * MAX_DET + it] = 0.0f;
        out[6 * MAX_DET + it] = 0.0f;
      }
    }
    __syncthreads();

    // ---- IoU suppression against the chosen box ----
    if (cok) {
      const float x1 = cb[0], y1 = cb[1], x2 = cb[2], y2 = cb[3], ar = cb[4];
      for (int j = tid; j < A; j += 1024) {
        float v = ms[j];
        if (!(v > NEG_INF)) continue;  // already dead: skip box read
        float4 b = *reinterpret_cast<const float4*>(boxes + (size_t)j * 4);
        float xx1 = fmaxf(x1, b.x), yy1 = fmaxf(y1, b.y);
        float xx2 = fminf(x2, b.z), yy2 = fminf(y2, b.w);
        float inter = fmaxf(xx2 - xx1, 0.0f) * fmaxf(yy2 - yy1, 0.0f);
        float iou = inter / (ar + areas[j] - inter);
        if (iou > NMS_THRESH) ms[j] = NEG_INF;
      }
    }
    __syncthreads();
  }
}

// ---------------------------------------------------------------------------
// Launch. Inputs: loc_pred [A,4] f32, cls_pred [A,64] f32, anchors [A,4] f32.
// Output (flat f32, 7000): boxes[1000*4] | labels[1000] | scores[1000] | valid[1000].
// Workspace layout (floats, Aal = A aligned to 1024):
//   boxes 4*Aal | areas Aal | scores Aal | labels Aal (int) | masked Aal
//   -> ~9.9 MB for A = 306,900.
// ---------------------------------------------------------------------------
extern "C" void kernel_launch(void* const* d_in, const int* in_sizes, int n_in,
                              void* d_out, int out_size, void* d_ws, size_t ws_size,
                              hipStream_t stream) {
  (void)n_in; (void)out_size; (void)ws_size;
  const float* loc = (const float*)d_in[0];
  const float* cls = (const float*)d_in[1];
  const float* anc = (const float*)d_in[2];
  const int A = in_sizes[0] / 4;
  const size_t Aal = ((size_t)A + 1023) & ~(size_t)1023;

  float* ws     = (float*)d_ws;
  float* boxes  = ws;
  float* areas  = boxes + 4 * Aal;
  float* scores = areas + Aal;
  int*   labels = (int*)(scores + Aal);
  float* masked = (float*)(labels + Aal);

  const int blocks = (A + 127) / 128;
  decode_score_kernel<<<blocks, 128, 0, stream>>>(loc, cls, anc, boxes, areas,
                                                  scores, labels, masked, A);
  nms_kernel<<<1, 1024, 0, stream>>>(boxes, areas, scores, labels, masked,
                                     (float*)d_out, A);
}